// SocialTrustGraphSAGE_2963527434318
// MI455X (gfx1250) — compile-verified
//
#include <hip/hip_runtime.h>

#define N_NODES 100000
#define N_EDGES 1280000
#define D 64
#define NTILES (N_NODES / 16)   // 6250, exact

typedef __attribute__((ext_vector_type(2))) float v2f;
typedef __attribute__((ext_vector_type(8))) float v8f;

// ---------------------------------------------------------------- zero
__global__ void zero_f32(float* __restrict__ p, int n) {
    int i = blockIdx.x * blockDim.x + threadIdx.x;
    if (i < n) p[i] = 0.0f;
}

// ---------------------------------------------------------------- degree
__global__ void degree_kernel(const long long* __restrict__ dst,
                              float* __restrict__ deg) {
    int e = blockIdx.x * blockDim.x + threadIdx.x;
    if (e < N_EDGES) {
        __hip_atomic_fetch_add(&deg[(int)dst[e]], 1.0f,
                               __ATOMIC_RELAXED, __HIP_MEMORY_SCOPE_AGENT);
    }
}

// ---------------------------------------------------------------- scatter-sum
// one thread per (edge, 4-feature chunk): E*16 threads, coalesced 256B/edge reads
__global__ void scatter_kernel(const float* __restrict__ x,
                               const long long* __restrict__ src,
                               const long long* __restrict__ dst,
                               float* __restrict__ agg) {
    int tid = blockIdx.x * blockDim.x + threadIdx.x;
    if (tid >= N_EDGES * 16) return;
    int e  = tid >> 4;
    int f4 = (tid & 15) * 4;
    int s = (int)src[e];
    int d = (int)dst[e];
    const float4 v = *(const float4*)(x + (long long)s * D + f4);
    float* base = agg + (long long)d * D + f4;
    __hip_atomic_fetch_add(base + 0, v.x, __ATOMIC_RELAXED, __HIP_MEMORY_SCOPE_AGENT);
    __hip_atomic_fetch_add(base + 1, v.y, __ATOMIC_RELAXED, __HIP_MEMORY_SCOPE_AGENT);
    __hip_atomic_fetch_add(base + 2, v.z, __ATOMIC_RELAXED, __HIP_MEMORY_SCOPE_AGENT);
    __hip_atomic_fetch_add(base + 3, v.w, __ATOMIC_RELAXED, __HIP_MEMORY_SCOPE_AGENT);
}

// ---------------------------------------------------------------- SAGE layer (WMMA f32)
// out[16x64 tile] = relu( (agg/deg) @ Wl^T + bl + xroot @ Wr^T )
// One wave32 per 16-node tile. fp32 WMMA 16x16x4, K=64 -> 16 k-steps,
// 4 column tiles, 2 matmuls fused into one accumulator: 128 wmma / wave.
__global__ __launch_bounds__(256)
void sage_wmma_kernel(const float* __restrict__ agg,
                      const float* __restrict__ deg,
                      const float* __restrict__ xroot,
                      const float* __restrict__ Wl,   // [D,D] row-major (out x in)
                      const float* __restrict__ bl,   // [D]
                      const float* __restrict__ Wr,   // [D,D]
                      float* __restrict__ out,        // [N,D]
                      int relu) {
    const int lane = threadIdx.x & 31;
    const int wave = threadIdx.x >> 5;
    const int tile = blockIdx.x * 8 + wave;
    if (tile >= NTILES) return;                 // wave-uniform: EXEC stays all-1s

    const int row   = tile * 16 + (lane & 15); // A-matrix row for this lane
    const int khalf = (lane >> 4) * 2;         // lanes 0-15 -> K=k0,k0+1; 16-31 -> k0+2,k0+3

    const float rdeg = 1.0f / fmaxf(deg[row], 1.0f);
    const float* aggr = agg   + (long long)row * D;
    const float* xr   = xroot + (long long)row * D;

    // A fragments for all 16 K-steps, both operands (mean and root)
    v2f a_mean[16], a_x[16];
#pragma unroll
    for (int kk = 0; kk < 16; ++kk) {
        const int k = 4 * kk + khalf;
        v2f am; am.x = aggr[k] * rdeg; am.y = aggr[k + 1] * rdeg;
        v2f ax; ax.x = xr[k];          ax.y = xr[k + 1];
        a_mean[kk] = am;
        a_x[kk]    = ax;
    }

    const int col = lane & 15;
#pragma unroll
    for (int n0 = 0; n0 < D; n0 += 16) {
        v8f c = {};
#pragma unroll
        for (int kk = 0; kk < 16; ++kk) {
            const int k = 4 * kk + khalf;
            // B[k][n] = W[n][k]  ->  (b.x,b.y) = W[n0+col][k], W[n0+col][k+1]
            const v2f bL = *(const v2f*)(Wl + (n0 + col) * D + k);
            const v2f bR = *(const v2f*)(Wr + (n0 + col) * D + k);
            c = __builtin_amdgcn_wmma_f32_16x16x4_f32(false, a_mean[kk], false, bL,
                                                      (short)0, c, false, false);
            c = __builtin_amdgcn_wmma_f32_16x16x4_f32(false, a_x[kk],    false, bR,
                                                      (short)0, c, false, false);
        }
        const float bias  = bl[n0 + col];
        const int   rbase = tile * 16 + ((lane >> 4) ? 8 : 0);
#pragma unroll
        for (int v = 0; v < 8; ++v) {
            float val = c[v] + bias;
            if (relu) val = fmaxf(val, 0.0f);
            out[(long long)(rbase + v) * D + n0 + col] = val;
        }
    }
}

// ---------------------------------------------------------------- head (64 -> 1)
__global__ void head_kernel(const float* __restrict__ h,
                            const float* __restrict__ Wh,
                            const float* __restrict__ bh,
                            float* __restrict__ out) {
    int n = blockIdx.x * blockDim.x + threadIdx.x;
    if (n >= N_NODES) return;
    const float* hr = h + (long long)n * D;
    float acc = 0.0f;
#pragma unroll
    for (int k = 0; k < D; k += 4) {
        float4 a = *(const float4*)(hr + k);
        float4 w = *(const float4*)(Wh + k);
        acc += a.x * w.x + a.y * w.y + a.z * w.z + a.w * w.w;
    }
    out[n] = acc + bh[0];
}

// ---------------------------------------------------------------- launch
extern "C" void kernel_launch(void* const* d_in, const int* in_sizes, int n_in,
                              void* d_out, int out_size, void* d_ws, size_t ws_size,
                              hipStream_t stream) {
    const float*     x   = (const float*)d_in[0];
    const long long* ei  = (const long long*)d_in[1];   // [2, E] int64
    const float*     W1l = (const float*)d_in[2];
    const float*     W1r = (const float*)d_in[3];
    const float*     b1  = (const float*)d_in[4];
    const float*     W2l = (const float*)d_in[5];
    const float*     W2r = (const float*)d_in[6];
    const float*     b2  = (const float*)d_in[7];
    const float*     Wh  = (const float*)d_in[8];
    const float*     bh  = (const float*)d_in[9];

    const long long* src = ei;
    const long long* dst = ei + N_EDGES;

    // workspace: deg [N] | agg [N*D] | h1 [N*D]  (~51.6 MB)
    float* deg = (float*)d_ws;
    float* agg = deg + N_NODES;
    float* h1  = agg + (size_t)N_NODES * D;

    // outputs: out [N] then h2 [N*D] (tuple return order)
    float* out_scalar = (float*)d_out;
    float* h2         = out_scalar + N_NODES;

    const int threads = 256;

    // zero deg + agg in one pass
    {
        int nz = N_NODES * (D + 1);
        zero_f32<<<(nz + threads - 1) / threads, threads, 0, stream>>>(deg, nz);
    }
    degree_kernel<<<(N_EDGES + threads - 1) / threads, threads, 0, stream>>>(dst, deg);

    // ---- layer 1
    {
        int nt = N_EDGES * 16;
        scatter_kernel<<<(nt + threads - 1) / threads, threads, 0, stream>>>(x, src, dst, agg);
    }
    sage_wmma_kernel<<<(NTILES + 7) / 8, threads, 0, stream>>>(agg, deg, x, W1l, b1, W1r, h1, 1);

    // ---- layer 2
    {
        int nz = N_NODES * D;
        zero_f32<<<(nz + threads - 1) / threads, threads, 0, stream>>>(agg, nz);
    }
    {
        int nt = N_EDGES * 16;
        scatter_kernel<<<(nt + threads - 1) / threads, threads, 0, stream>>>(h1, src, dst, agg);
    }
    sage_wmma_kernel<<<(NTILES + 7) / 8, threads, 0, stream>>>(agg, deg, h1, W2l, b2, W2r, h2, 1);

    // ---- head
    head_kernel<<<(N_NODES + threads - 1) / threads, threads, 0, stream>>>(h2, Wh, bh, out_scalar);
}